// HashEncoding_49890340110697
// MI455X (gfx1250) — compile-verified
//
#include <hip/hip_runtime.h>
#include <hip/hip_bf16.h>
#include <stdint.h>

// ---------------------------------------------------------------------------
// Instant-NGP hash-grid encoding for MI455X (gfx1250, wave32).
//   coords : [B=8, 3, N=65536]  float32
//   tables : [L=16, T=2^22, F=2] float32
//   out    : [B, L*F=32, N]      float32
// Gather-latency bound. Levers:
//   - 8 clause'd global_load_b64 gathers per level, level loop unrolled x2
//     (~16 outstanding loads/wave) for memory-level parallelism.
//   - global_prefetch_b8 software pipeline at distance 2 levels: address-only
//     prefetch of the next-next level's 8 corners (no VGPR cost for data).
//   - Non-temporal output stores + coord loads so the 64MB result / 6MB input
//     streams don't evict hot table lines from the 192MB L2.
// ---------------------------------------------------------------------------

#define NLEV   16
#define LOG2T  22
#define TMASK  ((1u << LOG2T) - 1u)
#define NPTS   65536
#define NB     8
#define P1     2654435761u
#define P2     805459861u
#define PFDIST 2          // prefetch distance in levels

// ---- Kernel 1: per-(batch, channel) min/max over N points --------------------
__global__ __launch_bounds__(256) void hg_minmax(const float* __restrict__ coords,
                                                 float* __restrict__ mm) {
  const int bc = blockIdx.x;                       // 0..23 = b*3+c
  const float4* __restrict__ p =
      (const float4*)(coords + (size_t)bc * NPTS); // 16384 float4 per (b,c)

  float4 v = p[threadIdx.x];
  float lmin = fminf(fminf(v.x, v.y), fminf(v.z, v.w));
  float lmax = fmaxf(fmaxf(v.x, v.y), fmaxf(v.z, v.w));
  #pragma unroll 2
  for (int i = threadIdx.x + 256; i < NPTS / 4; i += 256) {
    float4 u = p[i];
    lmin = fminf(lmin, fminf(fminf(u.x, u.y), fminf(u.z, u.w)));
    lmax = fmaxf(lmax, fmaxf(fmaxf(u.x, u.y), fmaxf(u.z, u.w)));
  }

  __shared__ float smin[256];
  __shared__ float smax[256];
  smin[threadIdx.x] = lmin;
  smax[threadIdx.x] = lmax;
  __syncthreads();
  #pragma unroll
  for (int s = 128; s > 0; s >>= 1) {
    if ((int)threadIdx.x < s) {
      smin[threadIdx.x] = fminf(smin[threadIdx.x], smin[threadIdx.x + s]);
      smax[threadIdx.x] = fmaxf(smax[threadIdx.x], smax[threadIdx.x + s]);
    }
    __syncthreads();
  }
  if (threadIdx.x == 0) {
    mm[bc * 2 + 0] = smin[0];
    mm[bc * 2 + 1] = smax[0];
  }
}

// Hash base for the (0,0,0) corner at a given resolution; corner k's hash is
// base + dx*1 + dy*P1 + dz*P2 (uint32 wraparound matches jnp.uint32 math).
__device__ __forceinline__ uint32_t corner_base(float nx, float ny, float nz,
                                                float res) {
  const float sx = nx * res, sy = ny * res, sz = nz * res;
  return (uint32_t)floorf(sx) + (uint32_t)floorf(sy) * P1 +
         (uint32_t)floorf(sz) * P2;
}

// ---- Kernel 2: hash-grid gather + trilinear interpolation --------------------
__global__ __launch_bounds__(256) void hg_encode(const float* __restrict__ coords,
                                                 const float2* __restrict__ tables,
                                                 const float* __restrict__ mm,
                                                 float* __restrict__ out) {
  const int n = blockIdx.x * 256 + threadIdx.x;    // point index
  const int b = blockIdx.y;                        // batch index

  // Streamed single-use coord loads: non-temporal ([B,3,N], stride N).
  const float* cb = coords + ((size_t)b * 3) * NPTS + (size_t)n;
  const float cx = __builtin_nontemporal_load(cb + 0 * NPTS);
  const float cy = __builtin_nontemporal_load(cb + 1 * NPTS);
  const float cz = __builtin_nontemporal_load(cb + 2 * NPTS);

  // Per-batch normalization constants (uniform within a block).
  const float mnx = mm[(b * 3 + 0) * 2 + 0], mxx = mm[(b * 3 + 0) * 2 + 1];
  const float mny = mm[(b * 3 + 1) * 2 + 0], mxy = mm[(b * 3 + 1) * 2 + 1];
  const float mnz = mm[(b * 3 + 2) * 2 + 0], mxz = mm[(b * 3 + 2) * 2 + 1];
  const float nx = (cx - mnx) / (mxx - mnx + 1e-6f);
  const float ny = (cy - mny) / (mxy - mny + 1e-6f);
  const float nz = (cz - mnz) / (mxz - mnz + 1e-6f);

  // Output base: out[b][2l+f][n], channel stride = NPTS (coalesced across lanes).
  float* __restrict__ op = out + ((size_t)b * (2 * NLEV)) * NPTS + (size_t)n;

  float res = 16.0f;
  #pragma unroll 2
  for (int l = 0; l < NLEV; ++l) {
    // ---- software prefetch pipeline: level l+PFDIST, address-only ----------
    if (l + PFDIST < NLEV) {
      const float2* __restrict__ tp = tables + ((size_t)(l + PFDIST) << LOG2T);
      const uint32_t bp = corner_base(nx, ny, nz, res * (float)(1 << PFDIST));
      __builtin_prefetch(tp + ((bp)                & TMASK), 0, 3);
      __builtin_prefetch(tp + ((bp + P2)           & TMASK), 0, 3);
      __builtin_prefetch(tp + ((bp + P1)           & TMASK), 0, 3);
      __builtin_prefetch(tp + ((bp + P1 + P2)      & TMASK), 0, 3);
      __builtin_prefetch(tp + ((bp + 1u)           & TMASK), 0, 3);
      __builtin_prefetch(tp + ((bp + 1u + P2)      & TMASK), 0, 3);
      __builtin_prefetch(tp + ((bp + 1u + P1)      & TMASK), 0, 3);
      __builtin_prefetch(tp + ((bp + 1u + P1 + P2) & TMASK), 0, 3);
    }

    // ---- gathers for level l ----------------------------------------------
    const float sx = nx * res, sy = ny * res, sz = nz * res;
    const float flx = floorf(sx), fly = floorf(sy), flz = floorf(sz);
    const float fx = sx - flx, fy = sy - fly, fz = sz - flz;
    const uint32_t base =
        (uint32_t)flx + (uint32_t)fly * P1 + (uint32_t)flz * P2;
    const float2* __restrict__ tab = tables + ((size_t)l << LOG2T);

    const float2 e0 = tab[(base)                & TMASK];   // (0,0,0)
    const float2 e1 = tab[(base + P2)           & TMASK];   // (0,0,1)
    const float2 e2 = tab[(base + P1)           & TMASK];   // (0,1,0)
    const float2 e3 = tab[(base + P1 + P2)      & TMASK];   // (0,1,1)
    const float2 e4 = tab[(base + 1u)           & TMASK];   // (1,0,0)
    const float2 e5 = tab[(base + 1u + P2)      & TMASK];   // (1,0,1)
    const float2 e6 = tab[(base + 1u + P1)      & TMASK];   // (1,1,0)
    const float2 e7 = tab[(base + 1u + P1 + P2) & TMASK];   // (1,1,1)

    // ---- trilinear weights + weighted sum ---------------------------------
    const float gx = 1.0f - fx, gy = 1.0f - fy, gz = 1.0f - fz;
    const float w0 = gx * gy * gz, w1 = gx * gy * fz;
    const float w2 = gx * fy * gz, w3 = gx * fy * fz;
    const float w4 = fx * gy * gz, w5 = fx * gy * fz;
    const float w6 = fx * fy * gz, w7 = fx * fy * fz;

    float o0 = w0 * e0.x;
    o0 = fmaf(w1, e1.x, o0); o0 = fmaf(w2, e2.x, o0); o0 = fmaf(w3, e3.x, o0);
    o0 = fmaf(w4, e4.x, o0); o0 = fmaf(w5, e5.x, o0); o0 = fmaf(w6, e6.x, o0);
    o0 = fmaf(w7, e7.x, o0);

    float o1 = w0 * e0.y;
    o1 = fmaf(w1, e1.y, o1); o1 = fmaf(w2, e2.y, o1); o1 = fmaf(w3, e3.y, o1);
    o1 = fmaf(w4, e4.y, o1); o1 = fmaf(w5, e5.y, o1); o1 = fmaf(w6, e6.y, o1);
    o1 = fmaf(w7, e7.y, o1);

    // Streaming output: non-temporal to preserve L2 for the hash tables.
    __builtin_nontemporal_store(o0, op);
    __builtin_nontemporal_store(o1, op + NPTS);
    op += 2 * NPTS;

    res *= 2.0f;
  }
}

// ---------------------------------------------------------------------------
extern "C" void kernel_launch(void* const* d_in, const int* in_sizes, int n_in,
                              void* d_out, int out_size, void* d_ws, size_t ws_size,
                              hipStream_t stream) {
  (void)in_sizes; (void)n_in; (void)out_size; (void)ws_size;

  const float*  coords = (const float*)d_in[0];   // [8,3,65536]
  const float2* tables = (const float2*)d_in[1];  // [16,2^22,2] viewed as float2
  float*        out    = (float*)d_out;           // [8,32,65536]
  float*        mm     = (float*)d_ws;            // 24 (min,max) pairs = 192 B

  hg_minmax<<<NB * 3, 256, 0, stream>>>(coords, mm);

  dim3 grid(NPTS / 256, NB);
  hg_encode<<<grid, 256, 0, stream>>>(coords, tables, mm, out);
}